// StructuredLogits_28802050687522
// MI455X (gfx1250) — compile-verified
//
#include <hip/hip_runtime.h>
#include <stdint.h>
#include <stddef.h>

#define BT 256            // B*T = 4*64
#define TILE 32

typedef __attribute__((ext_vector_type(2))) float v2f;
typedef __attribute__((ext_vector_type(8))) float v8f;

// ---------------------------------------------------------------------------
// CDNA5 async global->LDS helpers (gfx1250): GLOBAL_LOAD_ASYNC_TO_LDS_B128,
// tracked by ASYNCcnt, waited with S_WAIT_ASYNCCNT. LDS offset = low 32 bits
// of generic pointer into __shared__ (ISA 10.2 aperture mapping).
// ---------------------------------------------------------------------------
__device__ __forceinline__ void async_ld_b128(uint32_t lds_off, const float* gptr) {
    asm volatile("global_load_async_to_lds_b128 %0, %1, off"
                 :: "v"(lds_off), "v"(gptr) : "memory");
}
__device__ __forceinline__ void wait_async_le1() {
    asm volatile("s_wait_asynccnt 0x1" ::: "memory");
}
__device__ __forceinline__ void wait_async_0() {
    asm volatile("s_wait_asynccnt 0x0" ::: "memory");
}

// ---------------------------------------------------------------------------
// 1) per-row edge counts
// ---------------------------------------------------------------------------
__global__ void k_zero_i32(int* p, int n) {
    int i = blockIdx.x * blockDim.x + threadIdx.x;
    if (i < n) p[i] = 0;
}

__global__ void k_count(const int* __restrict__ row, int* __restrict__ cnt, int e) {
    int i = blockIdx.x * blockDim.x + threadIdx.x;
    if (i < e) atomicAdd(&cnt[row[i]], 1);
}

// ---------------------------------------------------------------------------
// 2) exclusive scan of cnt -> offsets (3 kernels; nb <= 1024 blocks assumed)
// ---------------------------------------------------------------------------
__global__ void k_scan_block(const int* __restrict__ cnt, int* __restrict__ offs,
                             int* __restrict__ bsum, int n) {
    __shared__ int s[256];
    int i = blockIdx.x * 256 + threadIdx.x;
    int v = (i < n) ? cnt[i] : 0;
    s[threadIdx.x] = v;
    __syncthreads();
    for (int d = 1; d < 256; d <<= 1) {
        int t = (threadIdx.x >= d) ? s[threadIdx.x - d] : 0;
        __syncthreads();
        s[threadIdx.x] += t;
        __syncthreads();
    }
    if (i < n) offs[i] = s[threadIdx.x] - v;        // block-local exclusive
    if (threadIdx.x == 255) bsum[blockIdx.x] = s[255];
}

__global__ void k_scan_top(const int* __restrict__ bsum, int* __restrict__ bpre, int nb) {
    __shared__ int s[1024];
    int t = threadIdx.x;
    int v = (t < nb) ? bsum[t] : 0;
    s[t] = v;
    __syncthreads();
    for (int d = 1; d < 1024; d <<= 1) {
        int x = (t >= d) ? s[t - d] : 0;
        __syncthreads();
        s[t] += x;
        __syncthreads();
    }
    if (t < nb) bpre[t] = s[t] - v;                  // exclusive
}

__global__ void k_scan_add(int* __restrict__ offs, const int* __restrict__ bpre,
                           int* __restrict__ cursor, int n) {
    int i = blockIdx.x * 256 + threadIdx.x;
    if (i < n) {
        int o = offs[i] + bpre[blockIdx.x];
        offs[i] = o;
        cursor[i] = o;
    }
}

// ---------------------------------------------------------------------------
// 3) CSR fill (atomic cursor) + deterministic per-row sort by edge id
// ---------------------------------------------------------------------------
__global__ void k_fill(const int* __restrict__ row, const int* __restrict__ col,
                       const float* __restrict__ vv, int* __restrict__ cursor,
                       int* __restrict__ ccol, float* __restrict__ cw,
                       int* __restrict__ ce, int e) {
    int i = blockIdx.x * blockDim.x + threadIdx.x;
    if (i < e) {
        int r = row[i];
        int p = atomicAdd(&cursor[r], 1);
        ccol[p] = col[i];
        cw[p]   = vv[i];
        ce[p]   = i;
    }
}

__global__ void k_sort_rows(const int* __restrict__ offs, const int* __restrict__ cnt,
                            int* __restrict__ ccol, float* __restrict__ cw,
                            int* __restrict__ ce, int n) {
    int j = blockIdx.x * blockDim.x + threadIdx.x;
    if (j >= n) return;
    int s = offs[j], d = cnt[j];
    for (int a = 0; a < d - 1; ++a) {             // selection sort, avg deg ~4
        int mi = a, me = ce[s + a];
        for (int b = a + 1; b < d; ++b) {
            int eb = ce[s + b];
            if (eb < me) { me = eb; mi = b; }
        }
        if (mi != a) {
            int t1 = ce[s + a];   ce[s + a] = ce[s + mi];   ce[s + mi] = t1;
            int t2 = ccol[s + a]; ccol[s + a] = ccol[s + mi]; ccol[s + mi] = t2;
            float t3 = cw[s + a]; cw[s + a] = cw[s + mi];   cw[s + mi] = t3;
        }
    }
}

// ---------------------------------------------------------------------------
// 4) transpose logits [BT][N] -> xT [N][BT]   (xT lives in d_out)
// ---------------------------------------------------------------------------
__global__ __launch_bounds__(256)
void k_transpose_in(const float* __restrict__ logits, float* __restrict__ xT, int n) {
    __shared__ float tile[TILE][TILE + 1];
    int j0 = blockIdx.x * TILE;     // N dim
    int b0 = blockIdx.y * TILE;     // BT dim
    for (int yy = threadIdx.y; yy < TILE; yy += 8) {
        int j = j0 + threadIdx.x, b = b0 + yy;
        tile[yy][threadIdx.x] = (j < n) ? logits[(size_t)b * n + j] : 0.f;
    }
    __syncthreads();
    for (int yy = threadIdx.y; yy < TILE; yy += 8) {
        int j = j0 + yy, b = b0 + threadIdx.x;
        if (j < n) xT[(size_t)j * BT + b] = tile[threadIdx.x][yy];
    }
}

// ---------------------------------------------------------------------------
// 5) heavy SpMM via WMMA: one block per 16 output rows.
//    ACC(16x256) += W(16xK) * X(Kx256), K = edges of the 16 rows (contiguous
//    CSR range). W columns are one-hot (edge -> local row), built in regs.
//    X rows (1KB each) are async-DMA'd into an LDS ring, 4 rows per chunk,
//    double buffered; each of the 8 waves owns two 16-col n-tiles and runs
//    V_WMMA_F32_16X16X4_F32 per 4-edge chunk.
// ---------------------------------------------------------------------------
__global__ __launch_bounds__(256)
void k_spmm16(const float* __restrict__ xT, const int* __restrict__ offs,
              const int* __restrict__ ccol, const float* __restrict__ cw,
              float* __restrict__ neighT, int n, int etot) {
    int j0 = blockIdx.x * 16;
    int t = threadIdx.x;
    int lane = t & 31, wave = t >> 5;

    __shared__ int   s_off[17];
    __shared__ int   s_col[256];
    __shared__ float s_w[256];
    __shared__ int   s_rid[256];
    __shared__ __align__(16) float xring[8][BT];   // 2 chunks x 4 rows x 1KB
    __shared__ float dump[16][BT];                 // ACC staging for coalesced store
    __shared__ float s_inv[16];

    if (t < 17) s_off[t] = (j0 + t < n) ? offs[j0 + t] : etot;
    __syncthreads();
    int start = s_off[0];
    int K = s_off[16] - start;

    v8f acc0 = {}; v8f acc1 = {};
    float wsum = 0.f;                    // thread r<16 accumulates row r's weight sum
    uint32_t ldsring = (uint32_t)(uintptr_t)(&xring[0][0]);

    for (int base = 0; base < K; base += 256) {
        int m = K - base; if (m > 256) m = 256;
        int mpad = (m + 3) & ~3;
        if (t < m) {
            int gi = start + base + t;
            s_col[t] = ccol[gi];
            s_w[t]   = cw[gi];
            int r = 0;                                // local row of this edge
            while (r < 15 && gi >= s_off[r + 1]) ++r;
            s_rid[t] = r;
        } else if (t < mpad) {
            s_col[t] = 0; s_w[t] = 0.f; s_rid[t] = 16;   // zero-weight padding
        }
        __syncthreads();
        if (t < 16) {                                  // deterministic row sums
            for (int k = 0; k < m; ++k)
                if (s_rid[k] == t) wsum += s_w[k];
        }
        int mc = mpad >> 2;
        // prologue: async-fetch chunk 0 (4 x-rows) into ring slot 0
        {
            int k = t >> 6;                            // row-in-chunk 0..3
            async_ld_b128(ldsring + (uint32_t)(k * 1024 + (t & 63) * 16),
                          xT + (size_t)s_col[k] * BT + (t & 63) * 4);
        }
        for (int c = 0; c < mc; ++c) {
            if (c + 1 < mc) {                          // prefetch next chunk
                int k = 4 * (c + 1) + (t >> 6);
                async_ld_b128(ldsring +
                              (uint32_t)((((c + 1) & 1) * 4 + (t >> 6)) * 1024 + (t & 63) * 16),
                              xT + (size_t)s_col[k] * BT + (t & 63) * 4);
                wait_async_le1();                      // chunk c landed (in-order)
            } else {
                wait_async_0();
            }
            __syncthreads();                           // publish ring slot c&1

            int kc = 4 * c;
            int mrow = lane & 15;
            int kb = kc + ((lane >> 4) << 1);          // A: VGPR0={K0,K2}, VGPR1={K1,K3}
            v2f a;
            a.x = (s_rid[kb]     == mrow) ? s_w[kb]     : 0.f;
            a.y = (s_rid[kb + 1] == mrow) ? s_w[kb + 1] : 0.f;
            const float* ring = &xring[(c & 1) * 4][0];
            int krel = (lane >> 4) << 1;               // B mirrors A's K split
            {
                int nb = (wave * 2 + 0) * 16 + (lane & 15);
                v2f b; b.x = ring[krel * BT + nb]; b.y = ring[(krel + 1) * BT + nb];
                acc0 = __builtin_amdgcn_wmma_f32_16x16x4_f32(
                           false, a, false, b, (short)0, acc0, false, false);
            }
            {
                int nb = (wave * 2 + 1) * 16 + (lane & 15);
                v2f b; b.x = ring[krel * BT + nb]; b.y = ring[(krel + 1) * BT + nb];
                acc1 = __builtin_amdgcn_wmma_f32_16x16x4_f32(
                           false, a, false, b, (short)0, acc1, false, false);
            }
            __syncthreads();                           // protect ring slot reuse
        }
    }

    // dump D fragments (VGPR r: lanes0-15 -> M=r, lanes16-31 -> M=r+8)
    {
        int mhalf = (lane >> 4) << 3;
        int n0 = (wave * 2 + 0) * 16 + (lane & 15);
        int n1 = (wave * 2 + 1) * 16 + (lane & 15);
        #pragma unroll
        for (int r = 0; r < 8; ++r) {
            dump[r + mhalf][n0] = acc0[r];
            dump[r + mhalf][n1] = acc1[r];
        }
    }
    if (t < 16) {
        int deg = s_off[t + 1] - s_off[t];
        s_inv[t] = (deg > 0) ? 1.0f / wsum : 0.f;
    }
    __syncthreads();
    #pragma unroll
    for (int r = 0; r < 16; ++r) {
        int j = j0 + r;
        if (j < n) neighT[(size_t)j * BT + t] = dump[r][t] * s_inv[r];
    }
}

// ---------------------------------------------------------------------------
// 6) out[bt][j] = neighT[j][bt] + logits[bt][j]   (tiled transpose + add)
// ---------------------------------------------------------------------------
__global__ __launch_bounds__(256)
void k_out(const float* __restrict__ neighT, const float* __restrict__ logits,
           float* __restrict__ out, int n) {
    __shared__ float tile[TILE][TILE + 1];
    int j0 = blockIdx.x * TILE;
    int b0 = blockIdx.y * TILE;
    for (int yy = threadIdx.y; yy < TILE; yy += 8) {
        int j = j0 + yy, b = b0 + threadIdx.x;
        tile[yy][threadIdx.x] = (j < n) ? neighT[(size_t)j * BT + b] : 0.f;
    }
    __syncthreads();
    for (int yy = threadIdx.y; yy < TILE; yy += 8) {
        int b = b0 + yy, j = j0 + threadIdx.x;
        if (j < n)
            out[(size_t)b * n + j] = tile[threadIdx.x][yy] + logits[(size_t)b * n + j];
    }
}

// ---------------------------------------------------------------------------
// launcher
// ---------------------------------------------------------------------------
static inline size_t align_up(size_t x, size_t a) { return (x + a - 1) & ~(a - 1); }

extern "C" void kernel_launch(void* const* d_in, const int* in_sizes, int n_in,
                              void* d_out, int out_size, void* d_ws, size_t ws_size,
                              hipStream_t stream) {
    const float* logits = (const float*)d_in[0];
    const float* vv     = (const float*)d_in[1];
    const int*   eidx   = (const int*)d_in[2];

    const int E = in_sizes[1];
    const int N = in_sizes[0] / BT;
    const int* row = eidx;
    const int* col = eidx + E;

    // workspace carve-up
    uint8_t* w = (uint8_t*)d_ws;
    size_t off = 0;
    auto take = [&](size_t bytes) -> void* {
        off = align_up(off, 256);
        void* p = w + off;
        off += bytes;
        return p;
    };
    int*   cnt     = (int*)take((size_t)N * 4);
    int*   offs    = (int*)take((size_t)N * 4);
    int*   cursor  = (int*)take((size_t)N * 4);
    int*   bsum    = (int*)take(1024 * 4);
    int*   bpre    = (int*)take(1024 * 4);
    int*   ccol    = (int*)take((size_t)E * 4);
    float* cw      = (float*)take((size_t)E * 4);
    int*   ce      = (int*)take((size_t)E * 4);
    float* neighT  = (float*)take((size_t)N * BT * 4);
    (void)ws_size;

    float* xT = (float*)d_out;   // reuse d_out as transposed-input scratch

    const int nbN = (N + 255) / 256;
    const int nbE = (E + 255) / 256;

    k_zero_i32<<<nbN, 256, 0, stream>>>(cnt, N);
    k_count<<<nbE, 256, 0, stream>>>(row, cnt, E);

    k_scan_block<<<nbN, 256, 0, stream>>>(cnt, offs, bsum, N);
    k_scan_top<<<1, 1024, 0, stream>>>(bsum, bpre, nbN);
    k_scan_add<<<nbN, 256, 0, stream>>>(offs, bpre, cursor, N);

    k_fill<<<nbE, 256, 0, stream>>>(row, col, vv, cursor, ccol, cw, ce, E);
    k_sort_rows<<<nbN, 256, 0, stream>>>(offs, cnt, ccol, cw, ce, N);

    dim3 tgrid((N + TILE - 1) / TILE, BT / TILE);
    dim3 tblk(TILE, 8);
    k_transpose_in<<<tgrid, tblk, 0, stream>>>(logits, xT, N);

    const int nb16 = (N + 15) / 16;
    k_spmm16<<<nb16, 256, 0, stream>>>(xT, offs, ccol, cw, neighT, N, E);

    k_out<<<tgrid, tblk, 0, stream>>>(neighT, logits, (float*)d_out, N);
}